// SpectralGraphConv_78142634983584
// MI455X (gfx1250) — compile-verified
//
#include <hip/hip_runtime.h>

#define NNODES 8192
#define DIM 128
#define KCHEB 3

typedef float v2f __attribute__((ext_vector_type(2)));
typedef float v8f __attribute__((ext_vector_type(8)));

__global__ void zero_kernel(float* __restrict__ p, int n) {
    int i = blockIdx.x * blockDim.x + threadIdx.x;
    if (i < n) p[i] = 0.0f;
}

// degree[r] += multiplicity (atomic float count; exact for counts < 2^24)
__global__ void degree_kernel(const long long* __restrict__ rows,
                              float* __restrict__ deg, int E) {
    int e = blockIdx.x * blockDim.x + threadIdx.x;
    if (e < E) atomicAdd(&deg[(int)rows[e]], 1.0f);
}

__global__ void dinv_kernel(float* __restrict__ deg, int n) {
    int i = blockIdx.x * blockDim.x + threadIdx.x;
    if (i < n) {
        float d = deg[i];
        deg[i] = d > 0.0f ? rsqrtf(d) : 0.0f;
    }
}

// y[r,:] += dinv[r]*dinv[c] * v[c,:]  — one wave per edge, float4 per lane
__global__ void spmm_kernel(const float* __restrict__ vin,
                            const long long* __restrict__ rows,
                            const long long* __restrict__ cols,
                            const float* __restrict__ dinv,
                            float* __restrict__ y, int E) {
    int gid = blockIdx.x * blockDim.x + threadIdx.x;
    int e = gid >> 5;
    int lane = gid & 31;
    if (e >= E) return;
    int r = (int)rows[e];
    int c = (int)cols[e];
    float s = dinv[r] * dinv[c];
    const float4* vc = (const float4*)&vin[c * DIM];
    float4 v = vc[lane];
    float* yr = &y[r * DIM + lane * 4];
    atomicAdd(yr + 0, s * v.x);
    atomicAdd(yr + 1, s * v.y);
    atomicAdd(yr + 2, s * v.z);
    atomicAdd(yr + 3, s * v.w);
}

// t1 = x - y ; rezero y for the second SpMM pass
__global__ void t1_kernel(const float* __restrict__ x, float* __restrict__ y,
                          float* __restrict__ t1, int n) {
    int i = blockIdx.x * blockDim.x + threadIdx.x;
    if (i < n) {
        t1[i] = x[i] - y[i];
        y[i] = 0.0f;
    }
}

// t2 = 2*(t1 - y) - x
__global__ void t2_kernel(const float* __restrict__ x, const float* __restrict__ t1,
                          const float* __restrict__ y, float* __restrict__ t2, int n) {
    int i = blockIdx.x * blockDim.x + threadIdx.x;
    if (i < n) t2[i] = 2.0f * (t1[i] - y[i]) - x[i];
}

// out = x@W0 + t1@W1 + t2@W2 + bias, via V_WMMA_F32_16X16X4_F32.
// 8 waves/block, one 16x16 output tile per wave; current 128x128 weight
// matrix staged in LDS (64KB) per Chebyshev term.
__global__ __launch_bounds__(256) void cheb_gemm_wmma(
    const float* __restrict__ x, const float* __restrict__ t1,
    const float* __restrict__ t2, const float* __restrict__ W,
    const float* __restrict__ bias, float* __restrict__ out) {
    extern __shared__ float smem[];  // DIM*DIM floats

    int wave = threadIdx.x >> 5;
    int lane = threadIdx.x & 31;
    int tile = blockIdx.x * 8 + wave;      // 0..4095
    int tn = tile & 7;                     // column tile (DIM/16 = 8)
    int tm = tile >> 3;                    // row tile (NNODES/16 = 512)
    int m  = lane & 15;                    // A row / B,C,D column within tile
    int hi = lane >> 4;                    // lane half selects K pair (A,B)
    int col = tn * 16 + m;

    v8f acc = {};
    for (int i = 0; i < KCHEB; ++i) {
        __syncthreads();
        for (int t = threadIdx.x; t < DIM * DIM; t += 256)
            smem[t] = W[i * DIM * DIM + t];
        __syncthreads();
        const float* A = (i == 0) ? x : (i == 1) ? t1 : t2;
        const float* arow = &A[(tm * 16 + m) * DIM];
#pragma unroll
        for (int kk = 0; kk < DIM / 4; ++kk) {
            int k = kk * 4 + 2 * hi;       // this lane's K pair within chunk
            v2f a = *(const v2f*)&arow[k]; // A[m][k], A[m][k+1]
            v2f b;
            b.x = smem[k * DIM + col];     // B[k][n]
            b.y = smem[(k + 1) * DIM + col];
            acc = __builtin_amdgcn_wmma_f32_16x16x4_f32(
                false, a, false, b, (short)0, acc, false, false);
        }
    }
    float bv = bias[col];
#pragma unroll
    for (int j = 0; j < 8; ++j) {
        int row = tm * 16 + j + 8 * hi;    // C/D layout: VGPR j, lane half
        out[row * DIM + col] = acc[j] + bv;
    }
}

extern "C" void kernel_launch(void* const* d_in, const int* in_sizes, int n_in,
                              void* d_out, int out_size, void* d_ws, size_t ws_size,
                              hipStream_t stream) {
    const float* x        = (const float*)d_in[0];
    const long long* ei   = (const long long*)d_in[1];
    const float* W        = (const float*)d_in[2];
    const float* bias     = (const float*)d_in[3];
    float* out            = (float*)d_out;

    int E = in_sizes[1] / 2;
    const long long* rows = ei;
    const long long* cols = ei + E;

    float* wsf  = (float*)d_ws;
    float* dinv = wsf;                       // N floats
    float* y    = wsf + NNODES;              // N*DIM
    float* t1   = y + NNODES * DIM;          // N*DIM
    float* t2   = t1 + NNODES * DIM;         // N*DIM

    int nzero = NNODES + NNODES * DIM;       // dinv + y (contiguous)
    zero_kernel<<<(nzero + 255) / 256, 256, 0, stream>>>(wsf, nzero);

    degree_kernel<<<(E + 255) / 256, 256, 0, stream>>>(rows, dinv, E);
    dinv_kernel<<<(NNODES + 255) / 256, 256, 0, stream>>>(dinv, NNODES);

    long long nth = (long long)E * 32;
    spmm_kernel<<<(int)((nth + 255) / 256), 256, 0, stream>>>(x, rows, cols, dinv, y, E);

    int nfeat = NNODES * DIM;
    t1_kernel<<<(nfeat + 255) / 256, 256, 0, stream>>>(x, y, t1, nfeat);

    spmm_kernel<<<(int)((nth + 255) / 256), 256, 0, stream>>>(t1, rows, cols, dinv, y, E);
    t2_kernel<<<(nfeat + 255) / 256, 256, 0, stream>>>(x, t1, y, t2, nfeat);

    int tiles = (NNODES / 16) * (DIM / 16);  // 4096
    cheb_gemm_wmma<<<tiles / 8, 256, DIM * DIM * sizeof(float), stream>>>(
        x, t1, t2, W, bias, out);
}